// TemporalGNN_9938554323266
// MI455X (gfx1250) — compile-verified
//
#include <hip/hip_runtime.h>
#include <math.h>

#define DIM_IN   4
#define PERIODS  12
#define OUTF     32
#define ASTRIDE  68   // padded LDS row stride (floats) for the [16 x 64] A tile

typedef float v2f __attribute__((ext_vector_type(2)));
typedef float v8f __attribute__((ext_vector_type(8)));

__device__ __forceinline__ void lds_fence() {
    asm volatile("s_wait_dscnt 0" ::: "memory");
}

__device__ __forceinline__ v8f splat8(float x) {
    v8f v;
#pragma unroll
    for (int i = 0; i < 8; ++i) v[i] = x;
    return v;
}

__device__ __forceinline__ v8f v8_sig(v8f x) {
    v8f y;
#pragma unroll
    for (int i = 0; i < 8; ++i) y[i] = 1.0f / (1.0f + __expf(-x[i]));
    return y;
}

__device__ __forceinline__ v8f v8_tanh(v8f x) {
    v8f y;
#pragma unroll
    for (int i = 0; i < 8; ++i) y[i] = tanhf(x[i]);
    return y;
}

__device__ __forceinline__ v8f v8_relu(v8f x) {
    v8f y;
#pragma unroll
    for (int i = 0; i < 8; ++i) y[i] = fmaxf(x[i], 0.0f);
    return y;
}

__device__ __forceinline__ v8f wmma_k4(v2f a, v2f b, v8f c) {
    // D = A(16x4,f32) x B(4x16,f32) + C(16x16,f32)
    return __builtin_amdgcn_wmma_f32_16x16x4_f32(false, a, false, b, (short)0, c,
                                                 false, false);
}

// Write a 16x32 C/D-layout register tile into row-major LDS at column base.
__device__ __forceinline__ void store_tile(float* A, int colbase, v8f t0, v8f t1,
                                           bool lo, int r) {
    const int rowoff = lo ? 0 : 8;
#pragma unroll
    for (int v = 0; v < 8; ++v) {
        A[(v + rowoff) * ASTRIDE + colbase + r]      = t0[v];
        A[(v + rowoff) * ASTRIDE + colbase + 16 + r] = t1[v];
    }
}

// [16x64] (LDS, row-major) @ Wl-half (packed B fragments in LDS) -> 16x16 tile.
__device__ __forceinline__ v8f gate_mm(const float* A, const float2* Wg, int lane,
                                       int r, int c0, int half, float bias) {
    v8f c = splat8(bias);
#pragma unroll
    for (int k = 0; k < 16; ++k) {
        v2f a = *(const v2f*)(A + r * ASTRIDE + 4 * k + c0);
        float2 bb = Wg[(k * 2 + half) * 32 + lane];
        v2f b = {bb.x, bb.y};
        c = wmma_k4(a, b, c);
    }
    return c;
}

// ---------------- Kernel 1: zero degree accumulator ----------------
__global__ void k_zero(float* p, int n) {
    int i = blockIdx.x * blockDim.x + threadIdx.x;
    if (i < n) p[i] = 0.0f;
}

// ---------------- Kernel 2: weighted in-degree ----------------
__global__ void k_deg(const int* dst, const float* ea, float* deg, int E) {
    int e = blockIdx.x * blockDim.x + threadIdx.x;
    if (e < E) atomicAdd(&deg[dst[e]], ea[e]);
}

// ---------------- Kernel 3: dis = rsqrt(deg+1); xprop self term ----------------
__global__ void k_self(const float* __restrict__ x, float* degdis,
                       float* __restrict__ xprop, int N) {
    int n = blockIdx.x * blockDim.x + threadIdx.x;
    if (n >= N) return;
    float d = degdis[n] + 1.0f;
    float dis = d > 0.0f ? rsqrtf(d) : 0.0f;
    degdis[n] = dis;
    float sn = dis * dis;
    const float* xn = x + (long)n * (DIM_IN * PERIODS);
    float* xp = xprop + (long)n * (DIM_IN * PERIODS);
#pragma unroll
    for (int dd = 0; dd < DIM_IN; ++dd)
#pragma unroll
        for (int t = 0; t < PERIODS; ++t)
            xp[t * DIM_IN + dd] = sn * xn[dd * PERIODS + t];  // [n][t][d] layout
}

// ---------------- Kernel 4: edge scatter of 4-dim features, all periods ----------------
__global__ void k_scatter(const float* __restrict__ x, const int* __restrict__ src,
                          const int* __restrict__ dst, const float* __restrict__ ea,
                          const float* __restrict__ dis, float* __restrict__ xprop,
                          int E) {
    int tid = blockIdx.x * blockDim.x + threadIdx.x;
    int e = tid >> 2;
    int dd = tid & 3;
    if (e >= E) return;
    int s = src[e], d_ = dst[e];
    float w = dis[s] * ea[e] * dis[d_];
    const float* xs = x + (long)s * (DIM_IN * PERIODS) + dd * PERIODS;
    float* xp = xprop + (long)d_ * (DIM_IN * PERIODS) + dd;
#pragma unroll
    for (int t = 0; t < PERIODS; ++t)
        atomicAdd(&xp[t * DIM_IN], w * xs[t]);
}

// ---------------- Kernel 5: WMMA GRU recurrence + output projection ----------------
__global__ __launch_bounds__(128) void k_gru(
    const float* __restrict__ xprop, const float* __restrict__ att,
    const float* __restrict__ Wc_z, const float* __restrict__ bc_z,
    const float* __restrict__ Wl_z, const float* __restrict__ bl_z,
    const float* __restrict__ Wc_r, const float* __restrict__ bc_r,
    const float* __restrict__ Wl_r, const float* __restrict__ bl_r,
    const float* __restrict__ Wc_h, const float* __restrict__ bc_h,
    const float* __restrict__ Wl_h, const float* __restrict__ bl_h,
    const float* __restrict__ W_out, const float* __restrict__ b_out,
    float* __restrict__ out, int N, int tiles) {
    __shared__ float2 Wpk[3 * 16 * 2 * 32];     // packed Wl B-fragments, 24 KB
    __shared__ float Abuf[4][16 * ASTRIDE];     // per-wave [16x64] tiles, padded

    const int tidb = threadIdx.x;

    // Stage Wl_{z,r,h} into LDS in exact B-fragment order:
    // entry (g,k,half,lane) = (Wl[4k+c0][half*16+lane%16], Wl[4k+c0+1][...]) with
    // c0 = (lane<16 ? 0 : 2).
    const float* Wls0 = Wl_z;
    const float* Wls1 = Wl_r;
    const float* Wls2 = Wl_h;
    for (int j = tidb; j < 3 * 16 * 2 * 32; j += blockDim.x) {
        int g = j >> 10;
        int rem = j & 1023;
        int k = rem >> 6;
        int half = (rem >> 5) & 1;
        int ln = rem & 31;
        int c0s = (ln < 16) ? 0 : 2;
        int col = half * 16 + (ln & 15);
        const float* W = (g == 0) ? Wls0 : ((g == 1) ? Wls1 : Wls2);
        float2 v;
        v.x = W[(4 * k + c0s) * OUTF + col];
        v.y = W[(4 * k + c0s + 1) * OUTF + col];
        Wpk[j] = v;
    }
    __syncthreads();

    const int wave = tidb >> 5;
    const int lane = tidb & 31;
    const int tile = blockIdx.x * 4 + wave;
    if (tile >= tiles) return;

    const int r = lane & 15;
    const bool lo = lane < 16;
    const int c0 = lo ? 0 : 2;
    float* A = &Abuf[wave][0];

    // softmax(attention) parameters (tiny, recomputed per thread)
    float m = att[0];
#pragma unroll
    for (int t = 1; t < PERIODS; ++t) m = fmaxf(m, att[t]);
    float s = 0.0f;
#pragma unroll
    for (int t = 0; t < PERIODS; ++t) s += __expf(att[t] - m);
    const float pinv = 1.0f / s;

    // Conv-weight B fragments (4x32 per gate) and biases, kept in registers.
    const int colh0 = r, colh1 = 16 + r;
    v2f bz0 = {Wc_z[c0 * OUTF + colh0], Wc_z[(c0 + 1) * OUTF + colh0]};
    v2f bz1 = {Wc_z[c0 * OUTF + colh1], Wc_z[(c0 + 1) * OUTF + colh1]};
    v2f br0 = {Wc_r[c0 * OUTF + colh0], Wc_r[(c0 + 1) * OUTF + colh0]};
    v2f br1 = {Wc_r[c0 * OUTF + colh1], Wc_r[(c0 + 1) * OUTF + colh1]};
    v2f bh0 = {Wc_h[c0 * OUTF + colh0], Wc_h[(c0 + 1) * OUTF + colh0]};
    v2f bh1 = {Wc_h[c0 * OUTF + colh1], Wc_h[(c0 + 1) * OUTF + colh1]};
    const float cbz0 = bc_z[colh0], cbz1 = bc_z[colh1];
    const float cbr0 = bc_r[colh0], cbr1 = bc_r[colh1];
    const float cbh0 = bc_h[colh0], cbh1 = bc_h[colh1];
    const float lbz0 = bl_z[colh0], lbz1 = bl_z[colh1];
    const float lbr0 = bl_r[colh0], lbr1 = bl_r[colh1];
    const float lbh0 = bl_h[colh0], lbh1 = bl_h[colh1];

    const int base = tile * 16;
    const int nodeA = min(base + r, N - 1);  // A-fragment source row

    v8f H0 = splat8(0.0f), H1 = splat8(0.0f);
    v8f acc0 = splat8(0.0f), acc1 = splat8(0.0f);

#pragma unroll 1
    for (int t = 0; t < PERIODS; ++t) {
        const float pt = __expf(att[t] - m) * pinv;
        // A fragment of propagated features (16x4), one float2 load per lane.
        v2f axp = *(const v2f*)(xprop + (long)nodeA * (DIM_IN * PERIODS) +
                                t * DIM_IN + c0);

        // --- Z gate ---
        v8f Gz0 = wmma_k4(axp, bz0, splat8(cbz0));
        v8f Gz1 = wmma_k4(axp, bz1, splat8(cbz1));
        store_tile(A, 0, Gz0, Gz1, lo, r);
        store_tile(A, 32, H0, H1, lo, r);
        lds_fence();
        v8f Z0 = v8_sig(gate_mm(A, Wpk, lane, r, c0, 0, lbz0));
        v8f Z1 = v8_sig(gate_mm(A, Wpk, lane, r, c0, 1, lbz1));

        // --- R gate ---
        v8f Gr0 = wmma_k4(axp, br0, splat8(cbr0));
        v8f Gr1 = wmma_k4(axp, br1, splat8(cbr1));
        store_tile(A, 0, Gr0, Gr1, lo, r);
        lds_fence();
        v8f R0 = v8_sig(gate_mm(A, Wpk + 1024, lane, r, c0, 0, lbr0));
        v8f R1 = v8_sig(gate_mm(A, Wpk + 1024, lane, r, c0, 1, lbr1));

        // --- candidate Ht ---
        v8f Gh0 = wmma_k4(axp, bh0, splat8(cbh0));
        v8f Gh1 = wmma_k4(axp, bh1, splat8(cbh1));
        store_tile(A, 0, Gh0, Gh1, lo, r);
        store_tile(A, 32, H0 * R0, H1 * R1, lo, r);
        lds_fence();
        v8f T0 = v8_tanh(gate_mm(A, Wpk + 2048, lane, r, c0, 0, lbh0));
        v8f T1 = v8_tanh(gate_mm(A, Wpk + 2048, lane, r, c0, 1, lbh1));

        H0 = Z0 * H0 + (1.0f - Z0) * T0;
        H1 = Z1 * H1 + (1.0f - Z1) * T1;
        acc0 += pt * H0;
        acc1 += pt * H1;
    }

    // Output projection: relu(acc)[16x32] @ W_out[32x12] (+ zero-padded cols).
    store_tile(A, 0, v8_relu(acc0), v8_relu(acc1), lo, r);
    lds_fence();
    v8f o = splat8(r < PERIODS ? b_out[r] : 0.0f);
#pragma unroll
    for (int k = 0; k < 8; ++k) {
        v2f a = *(const v2f*)(A + r * ASTRIDE + 4 * k + c0);
        int row0 = 4 * k + c0;
        v2f b = {r < PERIODS ? W_out[row0 * PERIODS + r] : 0.0f,
                 r < PERIODS ? W_out[(row0 + 1) * PERIODS + r] : 0.0f};
        o = wmma_k4(a, b, o);
    }
#pragma unroll
    for (int v = 0; v < 8; ++v) {
        int M = lo ? v : v + 8;
        int node = base + M;
        if (node < N && r < PERIODS) out[(long)node * PERIODS + r] = o[v];
    }
}

extern "C" void kernel_launch(void* const* d_in, const int* in_sizes, int n_in,
                              void* d_out, int out_size, void* d_ws, size_t ws_size,
                              hipStream_t stream) {
    const float* x    = (const float*)d_in[0];
    const int* eidx   = (const int*)d_in[1];
    const float* ea   = (const float*)d_in[2];
    const float* attn = (const float*)d_in[3];
    const float* Wc_z = (const float*)d_in[4];
    const float* bc_z = (const float*)d_in[5];
    const float* Wl_z = (const float*)d_in[6];
    const float* bl_z = (const float*)d_in[7];
    const float* Wc_r = (const float*)d_in[8];
    const float* bc_r = (const float*)d_in[9];
    const float* Wl_r = (const float*)d_in[10];
    const float* bl_r = (const float*)d_in[11];
    const float* Wc_h = (const float*)d_in[12];
    const float* bc_h = (const float*)d_in[13];
    const float* Wl_h = (const float*)d_in[14];
    const float* bl_h = (const float*)d_in[15];
    const float* Wout = (const float*)d_in[16];
    const float* bout = (const float*)d_in[17];
    float* out = (float*)d_out;

    const int N = in_sizes[0] / (DIM_IN * PERIODS);
    const int E = in_sizes[2];
    const int* src = eidx;
    const int* dst = eidx + E;

    // Workspace layout: deg/dis [N floats], xprop [N*48 floats]
    float* degdis = (float*)d_ws;
    size_t off = ((size_t)N * sizeof(float) + 255) & ~(size_t)255;
    float* xprop = (float*)((char*)d_ws + off);

    k_zero<<<(N + 255) / 256, 256, 0, stream>>>(degdis, N);
    k_deg<<<(E + 255) / 256, 256, 0, stream>>>(dst, ea, degdis, E);
    k_self<<<(N + 255) / 256, 256, 0, stream>>>(x, degdis, xprop, N);
    {
        long th = (long)E * 4;
        k_scatter<<<(unsigned)((th + 255) / 256), 256, 0, stream>>>(
            x, src, dst, ea, degdis, xprop, E);
    }
    const int tiles = (N + 15) / 16;
    k_gru<<<(tiles + 3) / 4, 128, 0, stream>>>(
        xprop, attn, Wc_z, bc_z, Wl_z, bl_z, Wc_r, bc_r, Wl_r, bl_r, Wc_h, bc_h,
        Wl_h, bl_h, Wout, bout, out, N, tiles);
}